// Operator_36988258353305
// MI455X (gfx1250) — compile-verified
//
#include <hip/hip_runtime.h>
#include <hip/hip_bf16.h>

// ---------------------------------------------------------------------------
// Spherical-harmonic spectral operator, MI455X (gfx1250, wave32, WMMA f32).
// Only the last time step flows through SHT -> channel-mix -> iSHT; the mean
// over time (xu) is the residual added at the end.
// ---------------------------------------------------------------------------

#define NLATc 361
#define NLONc 720
#define LMAXc 360
#define MMAXc 361
#define NCc   8
#define NBc   2
#define NTc   10
#define ROWSc (NBc * NCc * NLATc)     // 5776 = 361*16 (exact multiple of 16)
#define HWc   (NLATc * NLONc)         // 259920
#define M2c   (2 * MMAXc)             // 722 (re/im interleaved freq columns)
#define FTCOLS 736                    // 722 padded up to 46*16
#define KINVc 724                     // 722 padded up to 181*4
#define LEGLSTR (MMAXc * NLATc)       // 130321, stride of l in leg[l][m][j]

#define PI_D 3.14159265358979323846

typedef float v2f __attribute__((ext_vector_type(2)));
typedef float v8f __attribute__((ext_vector_type(8)));

__device__ inline v8f wmma_f32_4(v2f a, v2f b, v8f c) {
  // D = A(16x4 f32) * B(4x16 f32) + C(16x16 f32)
  return __builtin_amdgcn_wmma_f32_16x16x4_f32(false, a, false, b,
                                               (short)0, c, false, false);
}

// ---------------------------------------------------------------------------
// Kernel 0: build trig tables.
// fwdT[w][n] (stride FTCOLS): n=2m+ri -> (2pi/720)*{cos, -sin}(2pi*m*w/720);
//   columns 722..735 zero.
// invT[k][w] (stride 720):   k=2m+ri -> c_m*{cos, -sin}(2pi*m*w/720),
//   c_0=c_360=1 else 2; rows 722..723 zero.
// ---------------------------------------------------------------------------
__global__ void k_tables(float* __restrict__ fwdT, float* __restrict__ invT) {
  int i = blockIdx.x * blockDim.x + threadIdx.x;
  const int nFwd = NLONc * FTCOLS;
  const int nInv = KINVc * NLONc;
  if (i < nFwd) {
    int w = i / FTCOLS, n = i % FTCOLS;
    float v = 0.0f;
    if (n < M2c) {
      int m = n >> 1, ri = n & 1;
      int p = (int)(((long long)m * (long long)w) % NLONc);
      double ang = (double)p * (PI_D / 360.0);
      double s = 2.0 * PI_D / (double)NLONc;
      v = ri ? (float)(-sin(ang) * s) : (float)(cos(ang) * s);
    }
    fwdT[i] = v;
  } else if (i < nFwd + nInv) {
    int j = i - nFwd;
    int k = j / NLONc, w = j % NLONc;
    float v = 0.0f;
    if (k < M2c) {
      int m = k >> 1, ri = k & 1;
      double cm = (m == 0 || m == LMAXc) ? 1.0 : 2.0;
      int p = (int)(((long long)m * (long long)w) % NLONc);
      double ang = (double)p * (PI_D / 360.0);
      v = ri ? (float)(-sin(ang) * cm) : (float)(cos(ang) * cm);
    }
    invT[j] = v;
  }
}

// ---------------------------------------------------------------------------
// Kernel 1: temporal mean + last-step centering.
// xu[r*720+w] = mean_t x ; xc[r*720+w] = x[t=T-1] - xu ; r=(b*8+c)*361+h
// ---------------------------------------------------------------------------
__global__ void k_meansub(const float* __restrict__ x,
                          float* __restrict__ xu, float* __restrict__ xc) {
  int i = blockIdx.x * blockDim.x + threadIdx.x;
  if (i >= ROWSc * NLONc) return;
  int w = i % NLONc;
  int r = i / NLONc;
  int h = r % NLATc;
  int bc = r / NLATc;
  int b = bc / NCc, c = bc % NCc;
  size_t base = ((size_t)(b * NTc) * NCc + c) * HWc + (size_t)h * NLONc + w;
  const size_t tstr = (size_t)NCc * HWc;
  float s = 0.0f;
#pragma unroll
  for (int t = 0; t < NTc; ++t) s += x[base + (size_t)t * tstr];
  float mu = s * (1.0f / NTc);
  xu[i] = mu;
  xc[i] = x[base + (size_t)(NTc - 1) * tstr] - mu;
}

// ---------------------------------------------------------------------------
// Kernel 2: forward longitudinal DFT as WMMA GEMM.
// F[r][n] = sum_w xc[r][w] * fwdT[w][n],  r-tiles=361, n-tiles=46, K=720.
// ---------------------------------------------------------------------------
__global__ __launch_bounds__(32) void k_dft(const float* __restrict__ xc,
                                            const float* __restrict__ fwdT,
                                            float* __restrict__ F) {
  int mt = blockIdx.x;          // 0..360
  int nt = blockIdx.y;          // 0..45
  int lane = threadIdx.x;
  int half = lane >> 4;
  int l15 = lane & 15;
  int row = mt * 16 + l15;
  int n = nt * 16 + l15;
  const float* arow = xc + (size_t)row * NLONc;
  v8f acc = {0.f, 0.f, 0.f, 0.f, 0.f, 0.f, 0.f, 0.f};
#pragma unroll 2
  for (int kb = 0; kb < 180; ++kb) {
    int k = kb * 4 + half * 2;
    v2f a = {arow[k], arow[k + 1]};
    v2f b = {fwdT[(size_t)k * FTCOLS + n], fwdT[(size_t)(k + 1) * FTCOLS + n]};
    acc = wmma_f32_4(a, b, acc);
  }
#pragma unroll
  for (int p = 0; p < 8; ++p) {
    int orow = mt * 16 + p + half * 8;
    F[(size_t)orow * FTCOLS + n] = acc[p];
  }
}

// ---------------------------------------------------------------------------
// Kernel 3: Legendre analysis per m.
// xsht[m][l][col] = sum_j leg[l][m][j]*wq[j] * F[(col>>1)*361+j][2m+(col&1)]
// grid (m=361, ltile=23, ntile=2), K=361 (90 full steps + masked tail).
// ---------------------------------------------------------------------------
__global__ __launch_bounds__(32) void k_ana(const float* __restrict__ leg,
                                            const float* __restrict__ wq,
                                            const float* __restrict__ F,
                                            float* __restrict__ xsht) {
  int m = blockIdx.x, lt = blockIdx.y, nt = blockIdx.z;
  int lane = threadIdx.x;
  int half = lane >> 4;
  int l15 = lane & 15;
  int row = lt * 16 + l15;
  int rowc = row < (LMAXc - 1) ? row : (LMAXc - 1);  // clamp A loads
  int col = nt * 16 + l15;                            // 0..31
  int bc = col >> 1, ri = col & 1;
  const float* legm = leg + (size_t)rowc * LEGLSTR + (size_t)m * NLATc;
  const float* Fcol = F + (size_t)2 * m + ri;
  v8f acc = {0.f, 0.f, 0.f, 0.f, 0.f, 0.f, 0.f, 0.f};
#pragma unroll 2
  for (int kb = 0; kb < 90; ++kb) {
    int j = kb * 4 + half * 2;
    v2f a = {legm[j] * wq[j], legm[j + 1] * wq[j + 1]};
    v2f b = {Fcol[(size_t)(bc * NLATc + j) * FTCOLS],
             Fcol[(size_t)(bc * NLATc + j + 1) * FTCOLS]};
    acc = wmma_f32_4(a, b, acc);
  }
  // tail: j = 360 (only K-slot 0, i.e. lanes 0..15, contributes)
  {
    v2f a = {0.f, 0.f};
    v2f b = {0.f, 0.f};
    if (half == 0) {
      a.x = legm[NLATc - 1] * wq[NLATc - 1];
      b.x = Fcol[(size_t)(bc * NLATc + (NLATc - 1)) * FTCOLS];
    }
    acc = wmma_f32_4(a, b, acc);
  }
#pragma unroll
  for (int p = 0; p < 8; ++p) {
    int l = lt * 16 + p + half * 8;
    if (l < LMAXc) xsht[((size_t)m * LMAXc + l) * 32 + col] = acc[p];
  }
}

// ---------------------------------------------------------------------------
// Kernel 4: per-(l,m) 8x8 complex channel mix + residual (VALU).
// ysht[m][l][(b*8+C)*2+ri] = sum_c (A+iAi)[C][c][l][m]*x[c] + x[C]
// ---------------------------------------------------------------------------
__global__ void k_mix(const float* __restrict__ A, const float* __restrict__ Ai,
                      const float* __restrict__ xsht, float* __restrict__ ysht) {
  int i = blockIdx.x * blockDim.x + threadIdx.x;
  if (i >= MMAXc * LMAXc * NBc * NCc) return;
  int C = i & 7;
  int b = (i >> 3) & 1;
  int l = (i >> 4) % LMAXc;
  int m = i / (16 * LMAXc);
  size_t xb = ((size_t)m * LMAXc + l) * 32;
  float yr = 0.f, yi = 0.f;
#pragma unroll
  for (int c = 0; c < NCc; ++c) {
    size_t ao = ((size_t)(C * NCc + c) * LMAXc + l) * MMAXc + m;
    float ar = A[ao], aim = Ai[ao];
    float xr = xsht[xb + (b * NCc + c) * 2 + 0];
    float xi = xsht[xb + (b * NCc + c) * 2 + 1];
    yr += ar * xr - aim * xi;
    yi += ar * xi + aim * xr;
  }
  yr += xsht[xb + (b * NCc + C) * 2 + 0];
  yi += xsht[xb + (b * NCc + C) * 2 + 1];
  ysht[xb + (b * NCc + C) * 2 + 0] = yr;
  ysht[xb + (b * NCc + C) * 2 + 1] = yi;
}

// ---------------------------------------------------------------------------
// Kernel 5a: zero the K-pad columns (k = 722,723) of Fm.
// ---------------------------------------------------------------------------
__global__ void k_zeropad(float* __restrict__ Fm) {
  int i = blockIdx.x * blockDim.x + threadIdx.x;
  if (i >= ROWSc * 2) return;
  Fm[(size_t)(i >> 1) * KINVc + M2c + (i & 1)] = 0.0f;
}

// ---------------------------------------------------------------------------
// Kernel 5: Legendre synthesis per m.
// Fm[(col>>1)*361+j][2m+(col&1)] = sum_l leg[l][m][j] * ysht[m][l][col]
// grid (m=361, jtile=23, ntile=2), K=360 exact.
// ---------------------------------------------------------------------------
__global__ __launch_bounds__(32) void k_syn(const float* __restrict__ leg,
                                            const float* __restrict__ ysht,
                                            float* __restrict__ Fm) {
  int m = blockIdx.x, jt = blockIdx.y, nt = blockIdx.z;
  int lane = threadIdx.x;
  int half = lane >> 4;
  int l15 = lane & 15;
  int row = jt * 16 + l15;
  int rowc = row < (NLATc - 1) ? row : (NLATc - 1);  // clamp A loads
  int col = nt * 16 + l15;
  const float* legj = leg + (size_t)m * NLATc + rowc;  // + l*LEGLSTR
  const float* yb = ysht + (size_t)m * LMAXc * 32 + col;
  v8f acc = {0.f, 0.f, 0.f, 0.f, 0.f, 0.f, 0.f, 0.f};
#pragma unroll 2
  for (int kb = 0; kb < 90; ++kb) {
    int l0 = kb * 4 + half * 2;
    v2f a = {legj[(size_t)l0 * LEGLSTR], legj[(size_t)(l0 + 1) * LEGLSTR]};
    v2f b = {yb[(size_t)l0 * 32], yb[(size_t)(l0 + 1) * 32]};
    acc = wmma_f32_4(a, b, acc);
  }
#pragma unroll
  for (int p = 0; p < 8; ++p) {
    int j = jt * 16 + p + half * 8;
    if (j < NLATc)
      Fm[(size_t)((col >> 1) * NLATc + j) * KINVc + 2 * m + (col & 1)] = acc[p];
  }
}

// ---------------------------------------------------------------------------
// Kernel 6: inverse DFT + residual epilogue.
// out[r][w] = sum_k Fm[r][k] * invT[k][w] + xu[r][w]; K=724, N=720 exact.
// ---------------------------------------------------------------------------
__global__ __launch_bounds__(32) void k_idft(const float* __restrict__ Fm,
                                             const float* __restrict__ invT,
                                             const float* __restrict__ xu,
                                             float* __restrict__ out) {
  int rt = blockIdx.x;  // 0..360
  int wt = blockIdx.y;  // 0..44
  int lane = threadIdx.x;
  int half = lane >> 4;
  int l15 = lane & 15;
  int row = rt * 16 + l15;
  int w = wt * 16 + l15;
  const float* arow = Fm + (size_t)row * KINVc;
  v8f acc = {0.f, 0.f, 0.f, 0.f, 0.f, 0.f, 0.f, 0.f};
#pragma unroll 2
  for (int kb = 0; kb < 181; ++kb) {
    int k = kb * 4 + half * 2;
    v2f a = {arow[k], arow[k + 1]};
    v2f b = {invT[(size_t)k * NLONc + w], invT[(size_t)(k + 1) * NLONc + w]};
    acc = wmma_f32_4(a, b, acc);
  }
#pragma unroll
  for (int p = 0; p < 8; ++p) {
    int orow = rt * 16 + p + half * 8;
    size_t o = (size_t)orow * NLONc + (wt * 16 + l15);
    out[o] = acc[p] + xu[o];
  }
}

// ---------------------------------------------------------------------------
extern "C" void kernel_launch(void* const* d_in, const int* in_sizes, int n_in,
                              void* d_out, int out_size, void* d_ws, size_t ws_size,
                              hipStream_t stream) {
  (void)in_sizes; (void)n_in; (void)out_size; (void)ws_size;
  const float* x   = (const float*)d_in[0];
  const float* A   = (const float*)d_in[1];
  const float* Ai  = (const float*)d_in[2];
  const float* leg = (const float*)d_in[3];
  const float* wq  = (const float*)d_in[4];
  float* out = (float*)d_out;

  // workspace layout (floats); F/Fm share, xc/ysht share (sequenced on stream)
  float* ws = (float*)d_ws;
  const size_t nXU   = (size_t)ROWSc * NLONc;        // 4,158,720
  const size_t nFWD  = (size_t)NLONc * FTCOLS;       //   529,920
  const size_t nINV  = (size_t)KINVc * NLONc;        //   521,280
  const size_t nF    = (size_t)ROWSc * FTCOLS;       // 4,251,136 (>= ROWSc*KINVc)
  float* xu   = ws;
  float* xc   = xu + nXU;         // reused as ysht after k_dft consumes it
  float* ysht = xc;
  float* fwdT = xc + nXU;
  float* invT = fwdT + nFWD;
  float* F    = invT + nINV;      // reused as Fm after k_ana consumes it
  float* Fm   = F;
  float* xsht = F + nF;

  // 0) trig tables
  {
    int n = (int)(nFWD + nINV);
    k_tables<<<(n + 255) / 256, 256, 0, stream>>>(fwdT, invT);
  }
  // 1) temporal mean + centering of last step
  {
    int n = (int)nXU;
    k_meansub<<<(n + 255) / 256, 256, 0, stream>>>(x, xu, xc);
  }
  // 2) forward DFT (WMMA)
  k_dft<<<dim3(ROWSc / 16, FTCOLS / 16), 32, 0, stream>>>(xc, fwdT, F);
  // 3) Legendre analysis (WMMA)
  k_ana<<<dim3(MMAXc, 23, 2), 32, 0, stream>>>(leg, wq, F, xsht);
  // 4) channel mixing (writes ysht, aliases dead xc)
  {
    int n = MMAXc * LMAXc * NBc * NCc;
    k_mix<<<(n + 255) / 256, 256, 0, stream>>>(A, Ai, xsht, ysht);
  }
  // 5a) zero Fm pad columns (F is dead after k_ana)
  k_zeropad<<<(ROWSc * 2 + 255) / 256, 256, 0, stream>>>(Fm);
  // 5) Legendre synthesis (WMMA)
  k_syn<<<dim3(MMAXc, 23, 2), 32, 0, stream>>>(leg, ysht, Fm);
  // 6) inverse DFT + residual (WMMA)
  k_idft<<<dim3(ROWSc / 16, NLONc / 16), 32, 0, stream>>>(Fm, invT, xu, out);
}